// SLUTagging_25099788878439
// MI455X (gfx1250) — compile-verified
//
#include <hip/hip_runtime.h>
#include <hip/hip_bf16.h>
#include <stdint.h>

// Problem constants from the reference
#define B_DIM 64
#define S_DIM 512
#define W_DIM 256
#define H_DIM 1024

typedef __attribute__((ext_vector_type(4))) float v4f;

__global__ __launch_bounds__(256) void slu_blend_kernel(
    const float* __restrict__ out_word,        // [B, W, H]
    const float* __restrict__ out_char,        // [B, S, H]
    const int* __restrict__ word_idx,          // [B, S]
    const unsigned char* __restrict__ head,    // [B, S] (bool)
    const unsigned char* __restrict__ valid,   // [B, S] (bool)
    float* __restrict__ out)                   // [B, S, H]
{
    __shared__ v4f smem[256];                  // 4 KB: one H-row staged via async copy

    const int row = blockIdx.x;                // row = b*S + s
    const int b   = row >> 9;                  // S = 512
    const int tid = threadIdx.x;

    v4f* __restrict__ orow = (v4f*)(out + (size_t)row * H_DIM);

    // Positions past the sample's true length are zero in the reference.
    // d_out is poisoned, so we must write zeros explicitly (streaming store).
    if (!valid[row]) {
        v4f z = {0.0f, 0.0f, 0.0f, 0.0f};
        __builtin_nontemporal_store(z, orow + tid);
        return;
    }

    const int   widx  = word_idx[row];                 // block-uniform gather index
    const float rate  = head[row] ? 0.88f : 0.70f;     // block-uniform rate
    const float crate = 1.0f - rate;

    const v4f* __restrict__ wrow =
        (const v4f*)(out_word + ((size_t)b * W_DIM + (size_t)widx) * H_DIM);
    const v4f* __restrict__ crow =
        (const v4f*)(out_char + (size_t)row * H_DIM);

    v4f w, c;

#if defined(__gfx1250__)
    // Stage the gathered word row global->LDS via the CDNA5 async copy path
    // (ASYNCcnt-tracked), overlapping with the non-temporal char-row load.
    // Low 32 bits of a generic LDS pointer == LDS byte address (ISA 10.2).
    {
        uint32_t lds_addr = (uint32_t)(uintptr_t)&smem[tid];
        const v4f* gaddr  = wrow + tid;
        asm volatile("global_load_async_to_lds_b128 %0, %1, off"
                     :: "v"(lds_addr), "v"(gaddr)
                     : "memory");

        // Streaming char-row load in flight while the async copy runs.
        c = __builtin_nontemporal_load(crow + tid);

        asm volatile("s_wait_asynccnt 0" ::: "memory");
        w = smem[tid];                         // ds_load_b128, per-lane slot
    }
#else
    c = __builtin_nontemporal_load(crow + tid);
    w = wrow[tid];
#endif

    // feat = word * rate + char * (1 - rate)   (v_fma-friendly form)
    v4f r;
    r.x = __builtin_fmaf(w.x, rate, c.x * crate);
    r.y = __builtin_fmaf(w.y, rate, c.y * crate);
    r.z = __builtin_fmaf(w.z, rate, c.z * crate);
    r.w = __builtin_fmaf(w.w, rate, c.w * crate);

    // Output is write-once streaming: non-temporal store keeps L2 for word tiles.
    __builtin_nontemporal_store(r, orow + tid);
}

extern "C" void kernel_launch(void* const* d_in, const int* in_sizes, int n_in,
                              void* d_out, int out_size, void* d_ws, size_t ws_size,
                              hipStream_t stream) {
    const float* out_word         = (const float*)d_in[0];
    const float* out_char         = (const float*)d_in[1];
    const int* word_idx           = (const int*)d_in[2];
    const unsigned char* head     = (const unsigned char*)d_in[3];
    const unsigned char* valid    = (const unsigned char*)d_in[4];
    float* out                    = (float*)d_out;

    // One block per (b, s) row; 256 threads x float4 = 1024 = H elements.
    dim3 grid(B_DIM * S_DIM);
    dim3 block(256);
    slu_blend_kernel<<<grid, block, 0, stream>>>(out_word, out_char, word_idx,
                                                 head, valid, out);
}